// _MLPPairEdgeHead_87342454931782
// MI455X (gfx1250) — compile-verified
//
#include <hip/hip_runtime.h>
#include <hip/hip_bf16.h>
#include <math.h>

typedef float v2f __attribute__((ext_vector_type(2)));
typedef float v8f __attribute__((ext_vector_type(8)));

#define Bb 2
#define Nn 512
#define Ee 128
#define Hh 256
#define ROWS (Bb * Nn)          // 1024
#define LN_EPS 1e-5f

// ---------------------------------------------------------------------------
// Kernel 1: LayerNorm over E=128. One block (128 threads) per row.
// ---------------------------------------------------------------------------
__global__ void ln_kernel(const float* __restrict__ h,
                          const float* __restrict__ gamma,
                          const float* __restrict__ beta,
                          float* __restrict__ hn) {
    const int row = blockIdx.x;          // 0..1023
    const int t   = threadIdx.x;         // 0..127
    __shared__ float red[Ee];

    const float v = h[row * Ee + t];
    red[t] = v;
    __syncthreads();
    for (int s = Ee / 2; s > 0; s >>= 1) {
        if (t < s) red[t] += red[t + s];
        __syncthreads();
    }
    const float mu = red[0] * (1.0f / Ee);
    __syncthreads();

    const float d = v - mu;
    red[t] = d * d;
    __syncthreads();
    for (int s = Ee / 2; s > 0; s >>= 1) {
        if (t < s) red[t] += red[t + s];
        __syncthreads();
    }
    const float var = red[0] * (1.0f / Ee);
    const float r   = rsqrtf(var + LN_EPS);
    hn[row * Ee + t] = d * r * gamma[t] + beta[t];
}

// ---------------------------------------------------------------------------
// Kernel 2: fp32 WMMA GEMM (1024x128)@(128x256), one wave = one 16x16 tile.
// blockIdx.y == 0 : aP[m][n] = (hn·Wa)[m][n] + b1[n]            (row-major)
// blockIdx.y == 1 : bT[batch][n][j] = (hn·Wb)[m][n], m=batch*512+j (transposed)
// Uses V_WMMA_F32_16X16X4_F32 (K-loop of 32 steps).
// ---------------------------------------------------------------------------
__global__ void gemm_wmma_kernel(const float* __restrict__ hn,
                                 const float* __restrict__ W1,
                                 const float* __restrict__ b1,
                                 float* __restrict__ aP,
                                 float* __restrict__ bT) {
    const int tile = blockIdx.x;         // 0..1023 : 64 M-tiles x 16 N-tiles
    const int mt   = tile >> 4;
    const int nt   = tile & 15;
    const int which = blockIdx.y;        // 0 = Wa path, 1 = Wb path
    const float* __restrict__ W = W1 + (size_t)which * Ee * Hh; // Wa / Wb rows

    const int lane = threadIdx.x;        // 0..31 (wave32)
    const int half = lane >> 4;          // 0 | 1
    const int l16  = lane & 15;

    const int m = mt * 16 + l16;         // A-row for this lane
    const int n = nt * 16 + l16;         // B/D-col for this lane

    v8f acc = {0.f, 0.f, 0.f, 0.f, 0.f, 0.f, 0.f, 0.f};

    for (int k0 = 0; k0 < Ee; k0 += 4) {
        const int ka = k0 + 2 * half;
        v2f A, Bf;
        // A 16x4 fp32: VGPR0=K=ka, VGPR1=K=ka+1 (ISA 7.12.2, 32-bit A layout)
        A.x  = hn[m * Ee + ka];
        A.y  = hn[m * Ee + ka + 1];
        // B 4x16 fp32: lanes 0-15 hold K={k0,k0+1}, lanes 16-31 K={k0+2,k0+3}
        Bf.x = W[(size_t)ka * Hh + n];
        Bf.y = W[(size_t)(ka + 1) * Hh + n];
        acc = __builtin_amdgcn_wmma_f32_16x16x4_f32(
                  /*neg_a=*/false, A, /*neg_b=*/false, Bf,
                  /*c_mod=*/(short)0, acc, /*reuse_a=*/false, /*reuse_b=*/false);
    }

    if (which == 0) {
        const float bias = b1[n];
        #pragma unroll
        for (int r = 0; r < 8; ++r) {
            const int mo = mt * 16 + 8 * half + r;     // D row (M = 8*half + r)
            aP[(size_t)mo * Hh + n] = acc[r] + bias;
        }
    } else {
        #pragma unroll
        for (int r = 0; r < 8; ++r) {
            const int mo = mt * 16 + 8 * half + r;
            const int bb = mo >> 9;                    // batch
            const int j  = mo & (Nn - 1);
            bT[(size_t)bb * Hh * Nn + (size_t)n * Nn + j] = acc[r];
        }
    }
}

// ---------------------------------------------------------------------------
// Kernel 3 (hot): l[b,i,j] = b2 + sum_h w2[h] * gelu(aP[b,i,h] + bT[b,h,j])
// grid (2 j-halves, 512 i, 2 batches), block 256. a_i and w2 broadcast
// from LDS; bT streamed coalesced (L2-resident).
// ---------------------------------------------------------------------------
__global__ void pair_gelu_kernel(const float* __restrict__ aP,
                                 const float* __restrict__ bT,
                                 const float* __restrict__ w2,
                                 const float* __restrict__ b2,
                                 float* __restrict__ l) {
    const int j  = blockIdx.x * 256 + threadIdx.x;   // 0..511
    const int i  = blockIdx.y;                       // 0..511
    const int bb = blockIdx.z;                       // 0..1

    __shared__ float sA[Hh];
    __shared__ float sW[Hh];
    sA[threadIdx.x] = aP[((size_t)(bb * Nn + i)) * Hh + threadIdx.x]; // b1 folded
    sW[threadIdx.x] = w2[threadIdx.x];
    __syncthreads();

    const float* __restrict__ bp = bT + (size_t)bb * Hh * Nn + j;    // stride Nn per h
    float acc = b2[0];

    #pragma unroll 4
    for (int h = 0; h < Hh; ++h) {
        const float x = sA[h] + bp[(size_t)h * Nn];
        // exact GELU: 0.5*x*(1+erf(x/sqrt(2)))
        const float g = 0.5f * x * (1.0f + erff(x * 0.70710678118654752f));
        acc = fmaf(sW[h], g, acc);
    }
    l[((size_t)(bb * Nn + i)) * Nn + j] = acc;
}

// ---------------------------------------------------------------------------
// Kernel 4: out = 0.5*(l + l^T) per batch.
// ---------------------------------------------------------------------------
__global__ void symmetrize_kernel(const float* __restrict__ l,
                                  float* __restrict__ out) {
    const int idx = blockIdx.x * 256 + threadIdx.x;  // 0..524287
    const int bb  = idx >> 18;
    const int rem = idx & (Nn * Nn - 1);
    const int i   = rem >> 9;
    const int j   = rem & (Nn - 1);
    out[idx] = 0.5f * (l[idx] + l[(size_t)bb * Nn * Nn + (size_t)j * Nn + i]);
}

// ---------------------------------------------------------------------------
extern "C" void kernel_launch(void* const* d_in, const int* in_sizes, int n_in,
                              void* d_out, int out_size, void* d_ws, size_t ws_size,
                              hipStream_t stream) {
    const float* h     = (const float*)d_in[0];
    const float* gamma = (const float*)d_in[1];
    const float* beta  = (const float*)d_in[2];
    const float* W1    = (const float*)d_in[3];
    const float* b1    = (const float*)d_in[4];
    const float* w2    = (const float*)d_in[5];
    const float* b2    = (const float*)d_in[6];
    float* out = (float*)d_out;

    // Workspace layout (floats)
    float* ws  = (float*)d_ws;
    float* hn  = ws;                               // 1024*128  = 131072
    float* aP  = hn + (size_t)ROWS * Ee;           // 1024*256  = 262144
    float* bT  = aP + (size_t)ROWS * Hh;           // 2*256*512 = 262144
    float* lbuf= bT + (size_t)Bb * Hh * Nn;        // 2*512*512 = 524288

    // 1) LayerNorm
    ln_kernel<<<dim3(ROWS), dim3(Ee), 0, stream>>>(h, gamma, beta, hn);

    // 2) WMMA fp32 GEMMs: aP (with b1 folded) and bT (transposed)
    gemm_wmma_kernel<<<dim3(64 * 16, 2), dim3(32), 0, stream>>>(hn, W1, b1, aP, bT);

    // 3) Pairwise GELU reduction (dominant VALU work)
    pair_gelu_kernel<<<dim3(Nn / 256, Nn, Bb), dim3(256), 0, stream>>>(
        aP, bT, w2, b2, lbuf);

    // 4) Symmetrize
    symmetrize_kernel<<<dim3((Bb * Nn * Nn) / 256), dim3(256), 0, stream>>>(lbuf, out);
}